// BaseLayer_83425444757708
// MI455X (gfx1250) — compile-verified
//
#include <hip/hip_runtime.h>
#include <math.h>

typedef __attribute__((ext_vector_type(2))) float v2f;
typedef __attribute__((ext_vector_type(8))) float v8f;

#define N_IN   128
#define N_OUT  128
#define BATCH  512
#define K_DEG  3
#define NB     11          // G + K_DEG basis functions per input
#define NKNOT  15          // G + 2*K_DEG + 1 knots
#define KAUG   (N_IN * NB + N_IN)   // 1536 fused reduction dim

// ---------------------------------------------------------------------------
// Kernel 1: per-(batch, input) build augmented activation row:
//   A[b, i*11 + j] = B_j(x[b,i])   (cubic B-spline basis, Cox-de Boor)
//   A[b, 1408 + i] = silu(x[b,i])
// ---------------------------------------------------------------------------
__global__ __launch_bounds__(256) void kan_prep_act(
    const float* __restrict__ x, const float* __restrict__ grid,
    float* __restrict__ A) {
  int idx = blockIdx.x * blockDim.x + threadIdx.x;   // 0 .. BATCH*N_IN-1
  int b = idx >> 7;
  int i = idx & (N_IN - 1);
  float xv = x[idx];

  // All E rows of `grid` are identical; read the shared knot vector (row 0).
  float t[NKNOT];
#pragma unroll
  for (int m = 0; m < NKNOT; ++m) t[m] = grid[m];

  // Degree-0 indicators over the 14 knot spans.
  float bb[NKNOT - 1];
#pragma unroll
  for (int m = 0; m < NKNOT - 1; ++m)
    bb[m] = (xv >= t[m] && xv < t[m + 1]) ? 1.0f : 0.0f;

  // Cox-de Boor recursion up to degree 3 -> 11 basis values in bb[0..10].
#pragma unroll
  for (int kd = 1; kd <= K_DEG; ++kd) {
#pragma unroll
    for (int m = 0; m < NKNOT - 1 - kd; ++m) {
      float left  = (xv - t[m]) / (t[m + kd] - t[m]);
      float right = (t[m + kd + 1] - xv) / (t[m + kd + 1] - t[m + 1]);
      bb[m] = left * bb[m] + right * bb[m + 1];
    }
  }

  float* dst = A + b * KAUG + i * NB;
#pragma unroll
  for (int j = 0; j < NB; ++j) dst[j] = bb[j];

  // SiLU residual feature.
  A[b * KAUG + N_IN * NB + i] = xv / (1.0f + __expf(-xv));
}

// ---------------------------------------------------------------------------
// Kernel 2: fuse weights:
//   W[o, i*11 + j] = c_spl[o,i] * c_basis[o*128+i, j]
//   W[o, 1408 + i] = c_res[o,i]
// ---------------------------------------------------------------------------
__global__ __launch_bounds__(256) void kan_prep_w(
    const float* __restrict__ c_basis, const float* __restrict__ c_res,
    const float* __restrict__ c_spl, float* __restrict__ W) {
  int idx = blockIdx.x * blockDim.x + threadIdx.x;   // 0 .. N_OUT*N_IN-1
  int o = idx >> 7;
  int i = idx & (N_IN - 1);

  float cs = c_spl[idx];
  const float* cb = c_basis + idx * NB;              // e = o*128 + i = idx
  float* dst = W + o * KAUG + i * NB;
#pragma unroll
  for (int j = 0; j < NB; ++j) dst[j] = cs * cb[j];

  W[o * KAUG + N_IN * NB + i] = c_res[idx];
}

// ---------------------------------------------------------------------------
// Kernel 3: out[512,128] = A[512,1536] x W[128,1536]^T via V_WMMA_F32_16X16X4_F32
// One wave (32 lanes) per 16x16 output tile; 8 waves per block, 32 blocks.
//
// f32 WMMA operand layout (ISA 7.12.2):
//   A tile 16x4 : lanes 0-15 hold {K=0,K=1} of row M=lane, lanes 16-31 hold
//                 {K=2,K=3} of row M=lane-16  -> one contiguous float2 load
//   B tile 4x16 : mirrored (row K striped across lanes within a VGPR)
//   D tile 16x16: VGPR r, lanes 0-15 -> (M=r, N=lane); lanes 16-31 -> (M=r+8)
// ---------------------------------------------------------------------------
__global__ __launch_bounds__(256) void kan_gemm_wmma(
    const float* __restrict__ A, const float* __restrict__ W,
    float* __restrict__ out) {
  int wave = (blockIdx.x << 3) + (threadIdx.x >> 5);   // 0 .. 255
  int ot = (wave & 7) << 4;                            // output-col tile base
  int bt = (wave >> 3) << 4;                           // batch-row tile base

  int lane = threadIdx.x & 31;
  int half = lane >> 4;     // 0: K 0,1   1: K 2,3
  int l    = lane & 15;

  const float* arow = A + (size_t)(bt + l) * KAUG + (half << 1);
  const float* brow = W + (size_t)(ot + l) * KAUG + (half << 1);

  v8f acc = {};
#pragma unroll 8
  for (int k = 0; k < KAUG; k += 4) {
    v2f av = *(const v2f*)(arow + k);
    v2f bv = *(const v2f*)(brow + k);
    acc = __builtin_amdgcn_wmma_f32_16x16x4_f32(
        /*neg_a=*/false, av, /*neg_b=*/false, bv,
        /*c_mod=*/(short)0, acc, /*reuse_a=*/false, /*reuse_b=*/false);
  }

  float* obase = out + (size_t)(bt + (half << 3)) * N_OUT + ot + l;
#pragma unroll
  for (int r = 0; r < 8; ++r) obase[(size_t)r * N_OUT] = acc[r];
}

// ---------------------------------------------------------------------------
extern "C" void kernel_launch(void* const* d_in, const int* in_sizes, int n_in,
                              void* d_out, int out_size, void* d_ws, size_t ws_size,
                              hipStream_t stream) {
  const float* x       = (const float*)d_in[0];
  const float* grid    = (const float*)d_in[1];
  const float* c_basis = (const float*)d_in[2];
  const float* c_res   = (const float*)d_in[3];
  const float* c_spl   = (const float*)d_in[4];
  float* out = (float*)d_out;

  float* A = (float*)d_ws;                         // 512*1536 f32 = 3 MB
  float* W = A + (size_t)BATCH * KAUG;             // 128*1536 f32 = 768 KB

  kan_prep_act<<<(BATCH * N_IN) / 256, 256, 0, stream>>>(x, grid, A);
  kan_prep_w<<<(N_OUT * N_IN) / 256, 256, 0, stream>>>(c_basis, c_res, c_spl, W);
  kan_gemm_wmma<<<(BATCH / 16) * (N_OUT / 16) / 8, 256, 0, stream>>>(A, W, out);
}